// ForwardWarping_71657234366504
// MI455X (gfx1250) — compile-verified
//
#include <hip/hip_runtime.h>

typedef float v2f __attribute__((ext_vector_type(2)));
typedef float v8f __attribute__((ext_vector_type(8)));

#define BIGF 100000000.0f
#define FARF 1000000.0f
static constexpr int Hc  = 768;
static constexpr int Wc  = 1024;
static constexpr int HWc = Hc * Wc;

// -------------------------------------------------------------------------
// Per-batch combined projection matrix G (4x4):
//   [Px,Py,Pz,tdist]^T = G @ [u*d, v*d, d, 1]^T
// rows 0..2 = K[:3,:] @ T @ M4 ;  row 3 = row2 of (T @ M4),
// where M4 = [[inv(K[:3,:3]), 0],[0,0,0,1]].
// -------------------------------------------------------------------------
__global__ void fw_setup_G(const float* __restrict__ T, const float* __restrict__ K,
                           float* __restrict__ G, int B) {
  int b = threadIdx.x;
  if (b >= B) return;
  float a = K[0], bb = K[1], c = K[2];
  float d = K[4], e  = K[5], f = K[6];
  float g = K[8], h  = K[9], i = K[10];
  float det = a * (e * i - f * h) - bb * (d * i - f * g) + c * (d * h - e * g);
  float id = 1.0f / det;
  float ik[9] = {
    (e * i - f * h) * id, (c * h - bb * i) * id, (bb * f - c * e) * id,
    (f * g - d * i) * id, (a * i - c * g) * id,  (c * d - a * f) * id,
    (d * h - e * g) * id, (bb * g - a * h) * id, (a * e - bb * d) * id };
  const float* Tb = T + b * 16;
  float TM[16];
  for (int r = 0; r < 4; ++r) {
    for (int k2 = 0; k2 < 3; ++k2) {
      TM[r * 4 + k2] = Tb[r * 4 + 0] * ik[0 * 3 + k2]
                     + Tb[r * 4 + 1] * ik[1 * 3 + k2]
                     + Tb[r * 4 + 2] * ik[2 * 3 + k2];
    }
    TM[r * 4 + 3] = Tb[r * 4 + 3];
  }
  float* Gb = G + b * 16;
  for (int j = 0; j < 3; ++j)
    for (int k2 = 0; k2 < 4; ++k2)
      Gb[j * 4 + k2] = K[j * 4 + 0] * TM[0 * 4 + k2] + K[j * 4 + 1] * TM[1 * 4 + k2]
                     + K[j * 4 + 2] * TM[2 * 4 + k2] + K[j * 4 + 3] * TM[3 * 4 + k2];
  for (int k2 = 0; k2 < 4; ++k2) Gb[12 + k2] = TM[2 * 4 + k2];
}

__global__ void fw_init_zbuf(unsigned long long* __restrict__ zb, int n) {
  int t = blockIdx.x * blockDim.x + threadIdx.x;
  if (t < n) zb[t] = ~0ull;
}

// -------------------------------------------------------------------------
// Scatter: project every source pixel via V_WMMA_F32_16X16X4_F32 and z-test
// with a packed (depth_bits, src_idx) 64-bit atomicMin.
// A (16x4) = G (rows>=4 zero); B (4x16) = 16 pixel columns [ud,vd,d,1].
// D layout => lane n holds components 0..3 of its own pixel in D[0..3].
// Grid has no tail (nPix % 256 == 0) so EXEC is all-ones at the WMMAs.
// -------------------------------------------------------------------------
__global__ void fw_scatter(const float* __restrict__ depth,
                           const float* __restrict__ G,
                           unsigned long long* __restrict__ zbuf) {
  int tid  = blockIdx.x * blockDim.x + threadIdx.x;
  int lane = threadIdx.x & 31;
  int b    = tid / HWc;
  int pidx = tid - b * HWc;
  int v    = pidx / Wc;
  int u    = pidx - v * Wc;

  float dd = depth[tid];
  float ud = (float)u * dd;
  float vd = (float)v * dd;

  // A matrix: lanes 0-3 hold G[r][0],G[r][1]; lanes 16-19 hold G[r][2],G[r][3]
  const float* Gb = G + b * 16;
  int  arow = lane & 15;
  bool hi   = lane >= 16;
  float ax = 0.0f, ay = 0.0f;
  if (arow < 4) {
    int cb = hi ? 2 : 0;
    ax = Gb[arow * 4 + cb];
    ay = Gb[arow * 4 + cb + 1];
  }

  // B matrices: VGPR0 = {K0 lo-lanes, K2 hi-lanes}, VGPR1 = {K1 lo, K3 hi}
  float d_lo  = __shfl(dd, lane & 15);   // hi lanes: d of lane-16
  float ud_hi = __shfl(ud, lane | 16);   // lo lanes: ud of lane+16
  float vd_hi = __shfl(vd, lane | 16);

  v2f am; am.x = ax; am.y = ay;
  v2f b0; b0.x = hi ? d_lo : ud;  b0.y = hi ? 1.0f : vd;
  v2f b1; b1.x = hi ? dd : ud_hi; b1.y = hi ? 1.0f : vd_hi;
  v8f cz = {0.f,0.f,0.f,0.f,0.f,0.f,0.f,0.f};

  v8f r0 = __builtin_amdgcn_wmma_f32_16x16x4_f32(false, am, false, b0, (short)0, cz, false, false);
  v8f r1 = __builtin_amdgcn_wmma_f32_16x16x4_f32(false, am, false, b1, (short)0, cz, false, false);

  // lanes 0-15 own group0 pixels directly; lanes 16-31 pull group1 results
  float t0 = __shfl(r1[0], lane & 15);
  float t1 = __shfl(r1[1], lane & 15);
  float t2 = __shfl(r1[2], lane & 15);
  float t3 = __shfl(r1[3], lane & 15);
  float Px = hi ? t0 : r0[0];
  float Py = hi ? t1 : r0[1];
  float Pz = hi ? t2 : r0[2];
  float td = hi ? t3 : r0[3];

  if (td > 0.0f && td < FARF) {
    float den = Pz + 1e-7f;
    float up  = Px / den;
    float vp  = Py / den;
    float uif = fminf(fmaxf(rintf(up), 0.0f), (float)(Wc - 1));
    float vif = fminf(fmaxf(rintf(vp), 0.0f), (float)(Hc - 1));
    int   uv  = (int)vif * Wc + (int)uif;
    unsigned long long pack =
        ((unsigned long long)__float_as_uint(td) << 32) | (unsigned)pidx;
    atomicMin(zbuf + (size_t)b * HWc + uv, pack);
  }
}

// -------------------------------------------------------------------------
// Resolve z-buffer + hole-fill middle columns + emit nv_img/nv_depth/nv_mask
// -------------------------------------------------------------------------
__global__ void fw_resolve(const float* __restrict__ img,
                           const unsigned long long* __restrict__ zbuf,
                           float* __restrict__ out, int B, int nPix) {
  int o = blockIdx.x * blockDim.x + threadIdx.x;
  if (o >= nPix) return;
  int b    = o / HWc;
  int pidx = o - b * HWc;
  int v    = pidx / Wc;
  int u    = pidx - v * Wc;
  const unsigned long long* zb = zbuf + (size_t)b * HWc;

  unsigned long long z = zb[pidx];
  bool  bw = (z != ~0ull);
  float bd = bw ? __uint_as_float((unsigned)(z >> 32)) : BIGF;
  unsigned bs = (unsigned)z;

  const int lo = Wc / 4, hi2 = (Wc / 4) * 3;
  if (u >= lo && u < hi2) {
    // candidates in argmin priority: self, up, left, up-left (strict <)
    if (v >= 1) {
      unsigned long long z1 = zb[pidx - Wc];
      bool w1 = (z1 != ~0ull);
      float d1 = w1 ? __uint_as_float((unsigned)(z1 >> 32)) : BIGF;
      if (d1 < bd) { bd = d1; bs = (unsigned)z1; bw = w1; }
    }
    if (u - 1 >= lo) {
      unsigned long long z1 = zb[pidx - 1];
      bool w1 = (z1 != ~0ull);
      float d1 = w1 ? __uint_as_float((unsigned)(z1 >> 32)) : BIGF;
      if (d1 < bd) { bd = d1; bs = (unsigned)z1; bw = w1; }
    }
    if (v >= 1 && u - 1 >= lo) {
      unsigned long long z1 = zb[pidx - Wc - 1];
      bool w1 = (z1 != ~0ull);
      float d1 = w1 ? __uint_as_float((unsigned)(z1 >> 32)) : BIGF;
      if (d1 < bd) { bd = d1; bs = (unsigned)z1; bw = w1; }
    }
  }

  float r = 0.0f, g = 0.0f, bl = 0.0f;
  if (bw) {
    const float* ib = img + (size_t)b * 3 * HWc;
    r  = ib[bs];
    g  = ib[(size_t)HWc + bs];
    bl = ib[(size_t)2 * HWc + bs];
  }
  float m     = (bd == 0.0f) ? 1.0f : 0.0f;
  float inval = (bd > FARF)  ? 1.0f : 0.0f;
  float mOut  = (m + inval > 0.0f) ? 1.0f : 0.0f;

  size_t HWu  = (size_t)HWc;
  float* oimg = out;
  float* odep = out + (size_t)B * 3 * HWu;
  float* omsk = odep + (size_t)B * HWu;
  oimg[((size_t)b * 3 + 0) * HWu + pidx] = r;
  oimg[((size_t)b * 3 + 1) * HWu + pidx] = g;
  oimg[((size_t)b * 3 + 2) * HWu + pidx] = bl;
  odep[(size_t)b * HWu + pidx] = bd;
  omsk[(size_t)b * HWu + pidx] = mOut;
}

extern "C" void kernel_launch(void* const* d_in, const int* in_sizes, int n_in,
                              void* d_out, int out_size, void* d_ws, size_t ws_size,
                              hipStream_t stream) {
  const float* img   = (const float*)d_in[0];
  const float* depth = (const float*)d_in[1];
  const float* T     = (const float*)d_in[2];
  const float* K     = (const float*)d_in[3];
  int nPix = in_sizes[1];          // B*1*H*W
  int B    = nPix / HWc;

  float* G = (float*)d_ws;
  unsigned long long* zbuf = (unsigned long long*)((char*)d_ws + 1024);
  float* out = (float*)d_out;

  fw_setup_G<<<1, 32, 0, stream>>>(T, K, G, B);
  const int threads = 256;
  int blocks = (nPix + threads - 1) / threads;
  fw_init_zbuf<<<blocks, threads, 0, stream>>>(zbuf, nPix);
  fw_scatter<<<nPix / threads, threads, 0, stream>>>(depth, G, zbuf);
  fw_resolve<<<blocks, threads, 0, stream>>>(img, zbuf, out, B, nPix);
}